// DistMulti_5428838662772
// MI455X (gfx1250) — compile-verified
//
#include <hip/hip_runtime.h>

#define D   128
#define E   50000
#define S   16

typedef __attribute__((ext_vector_type(2))) float v2f;
typedef __attribute__((ext_vector_type(4))) float v4f;
typedef __attribute__((ext_vector_type(8))) float v8f;

// ---------------------------------------------------------------------------
// Phase 0: zero the 6 sum vectors in workspace (re-run every call: no state).
// ---------------------------------------------------------------------------
__global__ void zero_ws(float* __restrict__ ws) {
    int i = blockIdx.x * blockDim.x + threadIdx.x;
    if (i < 6 * D) ws[i] = 0.0f;
}

// ---------------------------------------------------------------------------
// Phase 1: six gathered row-sum reductions.
//   job 0: sum emb_A[ei0[1][:]]                      (pos0 tail sum)
//   job 1: sum emb_B[ei1[1][:]]                      (pos1 tail sum)
//   job 2: S * sum emb_A[nh0[:,0]]                   (neg-head m0)
//   job 3: S * sum emb_A[nh1[:,0]]                   (neg-head m1)
//   job 4: sum emb_A[nt0.flat]                       (neg-tail m0, E*S rows)
//   job 5: sum emb_B[nt1.flat]                       (neg-tail m1, E*S rows)
// One row per wave per iteration: lane l holds float4 at column 4l.
// ---------------------------------------------------------------------------
__global__ void sum_kernel(const float* __restrict__ embA, const float* __restrict__ embB,
                           const int* __restrict__ ei0, const int* __restrict__ ei1,
                           const int* __restrict__ nh0, const int* __restrict__ nh1,
                           const int* __restrict__ nt0, const int* __restrict__ nt1,
                           float* __restrict__ sums) {
    const int t = blockIdx.y;
    const float* emb; const int* idx; int count, stride; float scale;
    switch (t) {
        case 0:  emb = embA; idx = ei0 + E; count = E;     stride = 1; scale = 1.0f;       break;
        case 1:  emb = embB; idx = ei1 + E; count = E;     stride = 1; scale = 1.0f;       break;
        case 2:  emb = embA; idx = nh0;     count = E;     stride = S; scale = (float)S;   break;
        case 3:  emb = embA; idx = nh1;     count = E;     stride = S; scale = (float)S;   break;
        case 4:  emb = embA; idx = nt0;     count = E * S; stride = 1; scale = 1.0f;       break;
        default: emb = embB; idx = nt1;     count = E * S; stride = 1; scale = 1.0f;       break;
    }
    const int lane = threadIdx.x & 31;
    const int wave = threadIdx.x >> 5;
    const int wavesPerBlock = blockDim.x >> 5;
    const int gw   = blockIdx.x * wavesPerBlock + wave;
    const int gtot = gridDim.x * wavesPerBlock;

    v4f acc = {0.0f, 0.0f, 0.0f, 0.0f};
    for (int r = gw; r < count; r += gtot) {
        const int row = idx[r * stride];
        const v4f v = *(const v4f*)(emb + (long)row * D + lane * 4);
        acc += v;
    }

    __shared__ float lds[D];
    if (threadIdx.x < D) lds[threadIdx.x] = 0.0f;
    __syncthreads();
    atomicAdd(&lds[lane * 4 + 0], acc[0]);
    atomicAdd(&lds[lane * 4 + 1], acc[1]);
    atomicAdd(&lds[lane * 4 + 2], acc[2]);
    atomicAdd(&lds[lane * 4 + 3], acc[3]);
    __syncthreads();
    if (threadIdx.x < D)
        atomicAdd(&sums[t * D + threadIdx.x], lds[threadIdx.x] * scale);
}

// ---------------------------------------------------------------------------
// Phase 2: cmat[t] = K_{t&1} @ sums[t], batched as two 128x128 x 128x16 GEMMs
// via V_WMMA_F32_16X16X4_F32 (sum vectors occupy B columns 0..2, rest zero).
// 16 waves: wave = (m, row_tile). A layout per ISA 7.12.2 (16x4 f32 A tile):
//   lane L, vgpr j -> A[L%16][j + 2*(L/16)];  B: lane L, vgpr j -> B[j+2*(L/16)][L%16]
// C layout: vgpr r, lane L -> (M = r + 8*(L/16), N = L%16).
// EXEC stays all-ones through the WMMA loop (masking is branchless).
// ---------------------------------------------------------------------------
__global__ void matvec_wmma(const float* __restrict__ rel,
                            const float* __restrict__ sums,
                            float* __restrict__ cmat) {
    const int lane = threadIdx.x & 31;
    const int wave = threadIdx.x >> 5;      // 0..15
    const int m        = wave >> 3;         // relation 0/1
    const int row_base = (wave & 7) << 4;   // 0,16,...,112
    const int half = lane >> 4;             // 0 | 1
    const int n    = lane & 15;             // B/C column

    const float  bmask = (n < 3) ? 1.0f : 0.0f;
    const int    ncol  = (n < 3) ? n : 0;          // clamp so loads stay in bounds
    const float* srow  = sums + (m + 2 * ncol) * D;
    const float* Kbase = rel + m * D * D + (row_base + n) * D;  // row (row_base + M), M = lane%16

    v8f acc = {0.0f, 0.0f, 0.0f, 0.0f, 0.0f, 0.0f, 0.0f, 0.0f};
    for (int k0 = 0; k0 < D; k0 += 4) {
        const int kk = k0 + 2 * half;
        v2f a = *(const v2f*)(Kbase + kk);          // A[M][kk], A[M][kk+1]
        v2f b;
        b[0] = srow[kk + 0] * bmask;                // B[kk+0][n]
        b[1] = srow[kk + 1] * bmask;                // B[kk+1][n]
        acc = __builtin_amdgcn_wmma_f32_16x16x4_f32(
            /*neg_a=*/false, a, /*neg_b=*/false, b,
            /*c_mod=*/(short)0, acc, /*reuse_a=*/false, /*reuse_b=*/false);
    }
    if (n < 3) {
        const int tcol = m + 2 * n;                 // job id for this column
#pragma unroll
        for (int r = 0; r < 8; ++r)
            cmat[tcol * D + row_base + r + 8 * half] = acc[r];
    }
}

// ---------------------------------------------------------------------------
// Phase 3: gather-dot scoring. One row per wave per iteration; xor-shuffle
// reduction leaves the dot in all lanes, so the repeat-16 jobs (neg-tail)
// store 16 outputs coalesced from lanes 0..15.
// ---------------------------------------------------------------------------
__global__ void score_kernel(const float* __restrict__ embA, const float* __restrict__ embB,
                             const int* __restrict__ ei0, const int* __restrict__ ei1,
                             const int* __restrict__ nh0, const int* __restrict__ nh1,
                             const int* __restrict__ nt0, const int* __restrict__ nt1,
                             const float* __restrict__ cmat, float* __restrict__ out) {
    const int t = blockIdx.y;
    const float* emb; const int* idx; int count, stride, rep; long obase;
    switch (t) {
        case 0:  emb = embA; idx = ei0; count = E;     stride = 1; obase = 0;                       rep = 0; break;
        case 1:  emb = embA; idx = ei1; count = E;     stride = 1; obase = E;                       rep = 0; break;
        case 2:  emb = embA; idx = nh0; count = E * S; stride = 1; obase = 2L * E;                  rep = 0; break;
        case 3:  emb = embA; idx = nh1; count = E * S; stride = 1; obase = 2L * E + 1L * E * S;     rep = 0; break;
        case 4:  emb = embA; idx = nt0; count = E;     stride = S; obase = 2L * E + 2L * E * S;     rep = 1; break;
        default: emb = embB; idx = nt1; count = E;     stride = S; obase = 2L * E + 3L * E * S;     rep = 1; break;
    }
    const int lane = threadIdx.x & 31;
    const int wave = threadIdx.x >> 5;
    const int wavesPerBlock = blockDim.x >> 5;
    const int gw   = blockIdx.x * wavesPerBlock + wave;
    const int gtot = gridDim.x * wavesPerBlock;

    const v4f c = *(const v4f*)(cmat + t * D + lane * 4);  // c slice lives in registers

    for (int r = gw; r < count; r += gtot) {
        const int row = idx[r * stride];
        const v4f v = *(const v4f*)(emb + (long)row * D + lane * 4);
        float p = v[0] * c[0] + v[1] * c[1] + v[2] * c[2] + v[3] * c[3];
#pragma unroll
        for (int off = 16; off >= 1; off >>= 1)
            p += __shfl_xor(p, off, 32);
        if (rep) {
            if (lane < S) out[obase + (long)r * S + lane] = p;
        } else {
            if (lane == 0) out[obase + r] = p;
        }
    }
}

// ---------------------------------------------------------------------------
extern "C" void kernel_launch(void* const* d_in, const int* in_sizes, int n_in,
                              void* d_out, int out_size, void* d_ws, size_t ws_size,
                              hipStream_t stream) {
    const float* embA = (const float*)d_in[0];
    const float* embB = (const float*)d_in[1];
    const float* rel  = (const float*)d_in[2];
    const int*   ei0  = (const int*)d_in[3];
    const int*   ei1  = (const int*)d_in[4];
    const int*   nh0  = (const int*)d_in[5];
    const int*   nh1  = (const int*)d_in[6];
    const int*   nt0  = (const int*)d_in[7];
    const int*   nt1  = (const int*)d_in[8];
    float* out  = (float*)d_out;
    float* sums = (float*)d_ws;        // 6*128 floats
    float* cmat = sums + 6 * D;        // 6*128 floats

    zero_ws<<<3, 256, 0, stream>>>(sums);

    dim3 gsum(64, 6);
    sum_kernel<<<gsum, 256, 0, stream>>>(embA, embB, ei0, ei1, nh0, nh1, nt0, nt1, sums);

    matvec_wmma<<<1, 512, 0, stream>>>(rel, sums, cmat);

    dim3 gsc(256, 6);
    score_kernel<<<gsc, 256, 0, stream>>>(embA, embB, ei0, ei1, nh0, nh1, nt0, nt1, cmat, out);
}